// Actor_71141838291282
// MI455X (gfx1250) — compile-verified
//
#include <hip/hip_runtime.h>
#include <hip/hip_bf16.h>

typedef __attribute__((ext_vector_type(2))) float v2f;
typedef __attribute__((ext_vector_type(8))) float v8f;

#define D32 32
#define NGRAPH 64

__device__ __forceinline__ float seluf(float x) {
    const float alpha = 1.6732632423543772f;
    const float scale = 1.0507009873554805f;
    return x > 0.0f ? scale * x : scale * alpha * (__expf(x) - 1.0f);
}

__device__ __forceinline__ float sigmoidf_(float x) {
    return 1.0f / (1.0f + __expf(-x));
}

__device__ __forceinline__ float tanhf_(float x) {
    return 1.0f - 2.0f / (__expf(2.0f * x) + 1.0f);
}

// ---------------------------------------------------------------------------
// zero fill
// ---------------------------------------------------------------------------
__global__ void zero_kernel(float* __restrict__ p, long n) {
    long i = (long)blockIdx.x * blockDim.x + threadIdx.x;
    long stride = (long)gridDim.x * blockDim.x;
    for (; i < n; i += stride) p[i] = 0.0f;
}

// ---------------------------------------------------------------------------
// Message + scatter:  msg = selu([link[first], link[second]] @ Wm + bm)
//                     agg[second] += msg           (M edges, D=32)
// One wave handles 16 edges. GEMM tile: [16x64] @ [64x32] via
// V_WMMA_F32_16X16X4_F32 (16 k-steps x 2 n-tiles).
// Weights stored transposed in LDS so each B fragment is one aligned b64;
// rows padded to break stride-64 bank conflicts (64 banks).
// ---------------------------------------------------------------------------
#define MSG_WAVES 8
#define CATP 68              // 64 + 4 pad  (16B aligned, bank stride 4)
#define WMP  66              // 64 + 2 pad  (8B aligned,  bank stride 2)
__global__ void msg_kernel(const float* __restrict__ link,
                           const int*   __restrict__ first,
                           const int*   __restrict__ second,
                           const float* __restrict__ Wm,   // 64 x 32 row-major
                           const float* __restrict__ bm,   // 32
                           float*       __restrict__ agg,
                           int numGroups)
{
    __shared__ float wmT[32 * WMP];                 // WmT[n][k] = Wm[k][n]
    __shared__ float cat[MSG_WAVES][16][CATP];
    __shared__ int   sidx[MSG_WAVES][16];

    for (int i = threadIdx.x; i < 64 * 32; i += blockDim.x) {
        const int k = i >> 5;
        const int n = i & 31;
        wmT[n * WMP + k] = Wm[i];
    }
    __syncthreads();

    const int lane = threadIdx.x & 31;
    const int wid  = threadIdx.x >> 5;
    const int hi   = lane >> 4;       // 0: lanes 0-15, 1: lanes 16-31
    const int lm   = lane & 15;

    const float bias0 = bm[lm];
    const float bias1 = bm[16 + lm];

    const int waveStride = gridDim.x * MSG_WAVES;
    const int iters = (numGroups + waveStride - 1) / waveStride;

    for (int it = 0; it < iters; ++it) {
        const int g = blockIdx.x * MSG_WAVES + wid + it * waveStride;
        const bool active = g < numGroups;   // wave-uniform
        if (active) {
            const long e = (long)g * 16 + lm;
            const int idx = (hi == 0) ? first[e] : second[e];
            if (hi == 0) sidx[wid][lm] = second[e];
            const float4* src = (const float4*)(link + (long)idx * D32);
            float4* dst = (float4*)(&cat[wid][lm][hi * 32]);
#pragma unroll
            for (int j = 0; j < 8; ++j) dst[j] = src[j];
        }
        __syncthreads();
        if (active) {
            v8f c0 = {};
            v8f c1 = {};
#pragma unroll
            for (int kt = 0; kt < 16; ++kt) {
                const int k = kt * 4 + hi * 2;   // even -> 8B aligned
                const v2f a  = *(const v2f*)&cat[wid][lm][k];
                const v2f b0 = *(const v2f*)&wmT[lm * WMP + k];
                const v2f b1 = *(const v2f*)&wmT[(16 + lm) * WMP + k];
                c0 = __builtin_amdgcn_wmma_f32_16x16x4_f32(false, a, false, b0, (short)0, c0, false, false);
                c1 = __builtin_amdgcn_wmma_f32_16x16x4_f32(false, a, false, b1, (short)0, c1, false, false);
            }
            // C layout: element j -> (row = j + 8*hi, col = lm [+16 for c1])
#pragma unroll
            for (int j = 0; j < 8; ++j) {
                const int r  = j + 8 * hi;
                const int di = sidx[wid][r];
                float* aggRow = agg + (long)di * D32;
                unsafeAtomicAdd(aggRow + lm,      seluf(c0[j] + bias0));
                unsafeAtomicAdd(aggRow + 16 + lm, seluf(c1[j] + bias1));
            }
        }
        __syncthreads();
    }
}

// ---------------------------------------------------------------------------
// GRU cell over E rows:  xm = agg @ W + b0 ; hm = h @ U + b1 ; gates ; h'<-..
// One wave handles 16 rows. Two [16x32]@[32x96] GEMMs via WMMA f32 16x16x4:
// 8 k-steps x 6 n-tiles each. Gate columns (d, 32+d, 64+d) land in the same
// lane's accumulators, so the nonlinearity is register-local.
// ---------------------------------------------------------------------------
#define GRU_WAVES 8
#define XSP 36               // 32 + 4 pad (16B aligned, bank stride 36)
#define WTP 34               // 32 + 2 pad (8B aligned)
__global__ void gru_kernel(float*       __restrict__ link,   // h (in/out), E x 32
                           const float* __restrict__ agg,    // x, E x 32
                           const float* __restrict__ W,      // 32 x 96
                           const float* __restrict__ U,      // 32 x 96
                           const float* __restrict__ b,      // 2 x 96
                           int numGroups)
{
    __shared__ float wT[96 * WTP];                  // wT[c][k] = W[k][c]
    __shared__ float uT[96 * WTP];
    __shared__ float xs[GRU_WAVES][16][XSP];
    __shared__ float hs[GRU_WAVES][16][XSP];

    for (int i = threadIdx.x; i < 32 * 96; i += blockDim.x) {
        const int k = i / 96;
        const int c = i - k * 96;
        wT[c * WTP + k] = W[i];
        uT[c * WTP + k] = U[i];
    }
    __syncthreads();

    const int lane = threadIdx.x & 31;
    const int wid  = threadIdx.x >> 5;
    const int hi   = lane >> 4;
    const int lm   = lane & 15;

    // bias preload for the two gate columns this lane owns (d = lm, 16+lm)
    float b0g[2][3], b1g[2][3];
#pragma unroll
    for (int t = 0; t < 2; ++t) {
        const int d = lm + 16 * t;
#pragma unroll
        for (int q = 0; q < 3; ++q) {
            b0g[t][q] = b[q * 32 + d];
            b1g[t][q] = b[96 + q * 32 + d];
        }
    }

    const int waveStride = gridDim.x * GRU_WAVES;
    const int iters = (numGroups + waveStride - 1) / waveStride;

    for (int it = 0; it < iters; ++it) {
        const int g = blockIdx.x * GRU_WAVES + wid + it * waveStride;
        const bool active = g < numGroups;   // wave-uniform
        if (active) {
            const long row = (long)g * 16 + lm;
            if (hi == 0) {
                const float4* src = (const float4*)(agg + row * D32);
                float4* dst = (float4*)(&xs[wid][lm][0]);
#pragma unroll
                for (int j = 0; j < 8; ++j) dst[j] = src[j];
            } else {
                const float4* src = (const float4*)(link + row * D32);
                float4* dst = (float4*)(&hs[wid][lm][0]);
#pragma unroll
                for (int j = 0; j < 8; ++j) dst[j] = src[j];
            }
        }
        __syncthreads();
        if (active) {
            v8f xc[6] = {};
            v8f hc[6] = {};
#pragma unroll
            for (int kt = 0; kt < 8; ++kt) {
                const int k = kt * 4 + hi * 2;
                const v2f ax = *(const v2f*)&xs[wid][lm][k];
                const v2f ah = *(const v2f*)&hs[wid][lm][k];
#pragma unroll
                for (int nt = 0; nt < 6; ++nt) {
                    const v2f bw = *(const v2f*)&wT[(nt * 16 + lm) * WTP + k];
                    xc[nt] = __builtin_amdgcn_wmma_f32_16x16x4_f32(false, ax, false, bw, (short)0, xc[nt], false, false);
                    const v2f bu = *(const v2f*)&uT[(nt * 16 + lm) * WTP + k];
                    hc[nt] = __builtin_amdgcn_wmma_f32_16x16x4_f32(false, ah, false, bu, (short)0, hc[nt], false, false);
                }
            }
#pragma unroll
            for (int t = 0; t < 2; ++t) {
                const int d = lm + 16 * t;
#pragma unroll
                for (int j = 0; j < 8; ++j) {
                    const int r = j + 8 * hi;
                    const float xz = xc[0 + t][j] + b0g[t][0];
                    const float hz = hc[0 + t][j] + b1g[t][0];
                    const float xr = xc[2 + t][j] + b0g[t][1];
                    const float hr = hc[2 + t][j] + b1g[t][1];
                    const float xh = xc[4 + t][j] + b0g[t][2];
                    const float hh = hc[4 + t][j] + b1g[t][2];
                    const float z    = sigmoidf_(xz + hz);
                    const float rr   = sigmoidf_(xr + hr);
                    const float cand = tanhf_(xh + rr * hh);
                    const float hold = hs[wid][r][d];
                    link[((long)g * 16 + r) * D32 + d] = z * hold + (1.0f - z) * cand;
                }
            }
        }
        __syncthreads();
    }
}

// ---------------------------------------------------------------------------
// graph_state = segment_sum(link, graph_ids, NGRAPH)   (E x 32 -> 64 x 32)
// ---------------------------------------------------------------------------
__global__ void graph_segsum_kernel(const float* __restrict__ link,
                                    const int*   __restrict__ gid,
                                    float*       __restrict__ gs,
                                    int E)
{
    __shared__ float lgs[NGRAPH * D32];
    for (int i = threadIdx.x; i < NGRAPH * D32; i += blockDim.x) lgs[i] = 0.0f;
    __syncthreads();

    const int col = threadIdx.x & 31;
    const int ty  = threadIdx.x >> 5;   // 0..7
    const int rowsPerBlock = (E + gridDim.x - 1) / gridDim.x;
    const int r0 = blockIdx.x * rowsPerBlock;
    const int r1 = min(E, r0 + rowsPerBlock);

    for (int r = r0 + ty; r < r1; r += 8) {
        atomicAdd(&lgs[gid[r] * D32 + col], link[(long)r * D32 + col]);
    }
    __syncthreads();
    for (int i = threadIdx.x; i < NGRAPH * D32; i += blockDim.x) {
        unsafeAtomicAdd(&gs[i], lgs[i]);
    }
}

// ---------------------------------------------------------------------------
// Y[M x N] = act(X[M x K] @ Wt[K x N] + bias)   act: 0 = selu, 1 = relu
// ---------------------------------------------------------------------------
__global__ void dense_kernel(const float* __restrict__ X,
                             const float* __restrict__ Wt,
                             const float* __restrict__ bias,
                             float*       __restrict__ Y,
                             int Mrows, int K, int N, int act)
{
    const int idx = blockIdx.x * blockDim.x + threadIdx.x;
    if (idx >= Mrows * N) return;
    const int row = idx / N;
    const int col = idx - row * N;
    float acc = bias[col];
    const float* x = X + (long)row * K;
    for (int k = 0; k < K; ++k) acc += x[k] * Wt[(long)k * N + col];
    Y[idx] = (act == 1) ? fmaxf(acc, 0.0f) : seluf(acc);
}

// ---------------------------------------------------------------------------
extern "C" void kernel_launch(void* const* d_in, const int* in_sizes, int n_in,
                              void* d_out, int out_size, void* d_ws, size_t ws_size,
                              hipStream_t stream) {
    const int E = in_sizes[0] / D32;     // 262144
    const int M = in_sizes[1];           // 2097152

    const float* action = (const float*)d_in[0];
    const int*   first  = (const int*)d_in[1];
    const int*   second = (const int*)d_in[2];
    const int*   gid    = (const int*)d_in[3];
    const float* Wm  = (const float*)d_in[5];
    const float* bm  = (const float*)d_in[6];
    const float* W   = (const float*)d_in[7];
    const float* U   = (const float*)d_in[8];
    const float* b   = (const float*)d_in[9];
    const float* Wr1 = (const float*)d_in[10];
    const float* br1 = (const float*)d_in[11];
    const float* Wr2 = (const float*)d_in[12];
    const float* br2 = (const float*)d_in[13];
    const float* Wr3 = (const float*)d_in[14];
    const float* br3 = (const float*)d_in[15];

    float* ws   = (float*)d_ws;
    float* link = ws;                          // E*32
    float* agg  = link + (size_t)E * D32;      // E*32
    float* gs   = agg  + (size_t)E * D32;      // 64*32
    float* h1   = gs   + NGRAPH * D32;         // 64*256
    float* h2   = h1   + NGRAPH * 256;         // 64*256

    hipMemcpyAsync(link, action, (size_t)E * D32 * sizeof(float),
                   hipMemcpyDeviceToDevice, stream);

    const int msgGroups = M / 16;
    const int gruGroups = E / 16;

    for (int t = 0; t < 8; ++t) {
        zero_kernel<<<512, 256, 0, stream>>>(agg, (long)E * D32);
        msg_kernel<<<1024, 256, 0, stream>>>(link, first, second, Wm, bm, agg, msgGroups);
        gru_kernel<<<512, 256, 0, stream>>>(link, agg, W, U, b, gruGroups);
    }

    zero_kernel<<<8, 256, 0, stream>>>(gs, NGRAPH * D32);
    graph_segsum_kernel<<<256, 256, 0, stream>>>(link, gid, gs, E);

    dense_kernel<<<(NGRAPH * 256 + 255) / 256, 256, 0, stream>>>(gs, Wr1, br1, h1, NGRAPH, 32, 256, 0);
    dense_kernel<<<(NGRAPH * 256 + 255) / 256, 256, 0, stream>>>(h1, Wr2, br2, h2, NGRAPH, 256, 256, 0);
    dense_kernel<<<(NGRAPH * 256 + 255) / 256, 256, 0, stream>>>(h2, Wr3, br3, (float*)d_out, NGRAPH, 256, 256, 1);
}